// QuantumLayer_84696755077396
// MI455X (gfx1250) — compile-verified
//
#include <hip/hip_runtime.h>
#include <math.h>

#define NQ 4
#define DIM 16
#define NLAYERS 6

typedef __attribute__((ext_vector_type(2))) float v2f;
typedef __attribute__((ext_vector_type(8))) float v8f;

// ---------------------------------------------------------------------------
// Kernel 1: compose the 6 (Rot x4 + CNOT-ring) layers into one 16x16 complex
// unitary W via column simulation in LDS.  Store A = [Re W ; Im W] (32x16
// row-major floats) into workspace.  One block, 256 threads = 16 cols x 16 k.
// Runs once per launch; uses precise sincosf to keep the composed matrix at
// full f32 accuracy.
// ---------------------------------------------------------------------------
__global__ __launch_bounds__(256)
void build_unitary(const float* __restrict__ wts, float* __restrict__ A)
{
    __shared__ float2 Wm[DIM][DIM];            // [state row k][column]
    const int tid = threadIdx.x;
    const int col = tid >> 4;
    const int k   = tid & 15;

    Wm[k][col] = make_float2((k == col) ? 1.0f : 0.0f, 0.0f);
    __syncthreads();

    for (int l = 0; l < NLAYERS; ++l) {
        // ---- four Rot(phi,theta,omega) gates -------------------------------
        for (int w = 0; w < NQ; ++w) {
            const float phi   = wts[(l * NQ + w) * 3 + 0];
            const float theta = wts[(l * NQ + w) * 3 + 1];
            const float omega = wts[(l * NQ + w) * 3 + 2];
            float st, ct;  sincosf(0.5f * theta, &st, &ct);
            float sap, cap; sincosf(-0.5f * (phi + omega), &sap, &cap); // ep
            float sam, cam; sincosf(-0.5f * (phi - omega), &sam, &cam); // em
            // m00 = ep*c            m01 = -conj(em)*s
            // m10 = em*s            m11 =  conj(ep)*c
            const float2 m00 = make_float2( cap * ct,  sap * ct);
            const float2 m01 = make_float2(-cam * st,  sam * st);
            const float2 m10 = make_float2( cam * st,  sam * st);
            const float2 m11 = make_float2( cap * ct, -sap * ct);

            const int mask = 1 << (3 - w);
            const float2 lo = Wm[k & ~mask][col];
            const float2 hi = Wm[k |  mask][col];
            const float2 g0 = (k & mask) ? m10 : m00;
            const float2 g1 = (k & mask) ? m11 : m01;
            float2 nv;
            nv.x = g0.x * lo.x - g0.y * lo.y + g1.x * hi.x - g1.y * hi.y;
            nv.y = g0.x * lo.y + g0.y * lo.x + g1.x * hi.y + g1.y * hi.x;
            __syncthreads();
            Wm[k][col] = nv;
            __syncthreads();
        }
        // ---- CNOT ring: ctrl w -> tgt (w+r)%4, r = l%3 + 1 -----------------
        const int r = l % (NQ - 1) + 1;
        for (int w = 0; w < NQ; ++w) {
            const int c = w, t = (w + r) % NQ;
            const int src = ((k >> (3 - c)) & 1) ? (k ^ (1 << (3 - t))) : k;
            const float2 nv = Wm[src][col];
            __syncthreads();
            Wm[k][col] = nv;
            __syncthreads();
        }
    }

    A[k * 16 + col]        = Wm[k][col].x;   // rows 0..15  : Re W
    A[(16 + k) * 16 + col] = Wm[k][col].y;   // rows 16..31 : Im W
}

// ---------------------------------------------------------------------------
// Kernel 2: fused batch pass.  One wave handles a tile of 16 batch elements:
//   RY product state -> U = A*v via 8x V_WMMA_F32_16X16X4_F32
//   -> probs -> Z expectations -> coalesced float4 store.
//
// B-fragment element for chunk ch, lane L:  K = 4*ch + koff(L) + {0,1}
//   K bits [b0 b1 b2 b3]:  b0b1 = ch (compile-time), b2 = L>=16 (one select
//   of c2/s2 hoisted out of the loop), b3 -> .x uses c3 / .y uses s3.
// Half-angles are in [0, pi], so hardware v_sin/v_cos (__sincosf) are
// accurate to a few ulp; error is further damped by products and squares.
// ---------------------------------------------------------------------------
__global__ __launch_bounds__(256)
void qsim_wmma(const float* __restrict__ x, const float* __restrict__ A,
               float* __restrict__ out, int nTiles)
{
    const int  lane = threadIdx.x & 31;
    const int  tile = blockIdx.x * (blockDim.x >> 5) + (threadIdx.x >> 5);
    if (tile >= nTiles) return;                 // wave-uniform: EXEC stays full

    const int  n      = lane & 15;              // batch element within tile (N)
    const int  batch  = tile * 16 + n;
    const bool hiHalf = (lane >= 16);

    // ---- half-angle factors of the RY product state ------------------------
    const float4 xx = reinterpret_cast<const float4*>(x)[batch];
    float s0, c0, s1, c1, s2, c2, s3, c3;
    __sincosf(0.5f * xx.x, &s0, &c0);
    __sincosf(0.5f * xx.y, &s1, &c1);
    __sincosf(0.5f * xx.z, &s2, &c2);
    __sincosf(0.5f * xx.w, &s3, &c3);

    const float t2 = hiHalf ? s2 : c2;          // only lane-dependent select
    const float q01[4] = { c0 * c1, c0 * s1, s0 * c1, s0 * s1 };

    // ---- A fragments: lanes 0-15 -> {K+0,K+1}, lanes 16-31 -> {K+2,K+3} ----
    const float* Are = A + n * 16 + (hiHalf ? 2 : 0);
    const float* Aim = Are + 16 * 16;

    v8f dre  = {};                              // Re rows of A (M = 0..15)
    v8f dim2 = {};                              // Im rows of A

    #pragma unroll
    for (int ch = 0; ch < 4; ++ch) {
        v2f aRe, aIm, b;
        aRe.x = Are[4 * ch];  aRe.y = Are[4 * ch + 1];
        aIm.x = Aim[4 * ch];  aIm.y = Aim[4 * ch + 1];
        const float pref = q01[ch] * t2;        // b0b1 * b2 factors
        b.x = pref * c3;                        // b3 = 0
        b.y = pref * s3;                        // b3 = 1
        dre  = __builtin_amdgcn_wmma_f32_16x16x4_f32(false, aRe, false, b,
                                                     (short)0, dre,  false, false);
        dim2 = __builtin_amdgcn_wmma_f32_16x16x4_f32(false, aIm, false, b,
                                                     (short)0, dim2, false, false);
    }

    // ---- probs + Z-sign reduction ------------------------------------------
    // C/D layout: lane holds N = lane&15; VGPR r holds M = r (+8 for lanes>=16)
    // m = mbase + r with mbase in {0,8}: qubit-0 sign = whole-half flip of psum,
    // qubit 1..3 signs depend only on compile-time r bits.
    float psum = 0.f, a1 = 0.f, a2 = 0.f, a3 = 0.f;
    #pragma unroll
    for (int r = 0; r < 8; ++r) {
        const float p = dre[r] * dre[r] + dim2[r] * dim2[r];
        psum += p;
        a1 += (r & 4) ? -p : p;
        a2 += (r & 2) ? -p : p;
        a3 += (r & 1) ? -p : p;
    }
    float a0 = hiHalf ? -psum : psum;

    // lane L and L^16 hold complementary M-halves of the same batch element
    a0 += __shfl_xor(a0, 16, 32);
    a1 += __shfl_xor(a1, 16, 32);
    a2 += __shfl_xor(a2, 16, 32);
    a3 += __shfl_xor(a3, 16, 32);

    if (!hiHalf) {
        float4 o; o.x = a0; o.y = a1; o.z = a2; o.w = a3;
        reinterpret_cast<float4*>(out)[batch] = o;
    }
}

// ---------------------------------------------------------------------------
extern "C" void kernel_launch(void* const* d_in, const int* in_sizes, int n_in,
                              void* d_out, int out_size, void* d_ws, size_t ws_size,
                              hipStream_t stream)
{
    const float* x   = (const float*)d_in[0];   // (BATCH, 4) float32
    const float* wts = (const float*)d_in[1];   // (6, 4, 3)  float32
    float*       A   = (float*)d_ws;            // 32x16 floats (2 KB)
    float*       out = (float*)d_out;           // (BATCH, 4) float32

    const int batchN = in_sizes[0] / NQ;        // 262144
    const int nTiles = batchN / 16;             // 16384 wave-tiles
    const int wavesPerBlock = 8;                // 256 threads, wave32
    const int blocks = (nTiles + wavesPerBlock - 1) / wavesPerBlock;

    build_unitary<<<1, 256, 0, stream>>>(wts, A);
    qsim_wmma<<<blocks, 256, 0, stream>>>(x, A, out, nTiles);
}